// KGCN_49959059587727
// MI455X (gfx1250) — compile-verified
//
#include <hip/hip_runtime.h>

#define DIM 32
#define NN  32
#define NREL 64
#define BATCH 1024

typedef float v2f __attribute__((ext_vector_type(2)));
typedef float v8f __attribute__((ext_vector_type(8)));

__device__ __forceinline__ float wave_max(float v) {
#pragma unroll
  for (int o = 16; o; o >>= 1) v = fmaxf(v, __shfl_xor(v, o, 32));
  return v;
}
__device__ __forceinline__ float wave_sum(float v) {
#pragma unroll
  for (int o = 16; o; o >>= 1) v += __shfl_xor(v, o, 32);
  return v;
}

// One 16x16 tile of X(16x32, LDS row-major stride 32) @ W(32x32, LDS) using
// 8 chained V_WMMA_F32_16X16X4_F32 (K=4 each, K total = 32).
// f32 A 16x4 layout: lane -> M = lane&15, VGPR j -> K = 2*(lane>>4) + j
// f32 B 4x16 layout: lane -> N = lane&15, VGPR j -> K = 2*(lane>>4) + j
__device__ __forceinline__ v8f wmma_x_w(const float* __restrict__ XL,
                                        const float* __restrict__ WL,
                                        int tn, int lane) {
  const int lrow = lane & 15;
  const int half = lane >> 4;
  v8f c = {};
#pragma unroll
  for (int k0 = 0; k0 < DIM; k0 += 4) {
    v2f a, bb;
    a.x  = XL[lrow * DIM + k0 + 2 * half + 0];
    a.y  = XL[lrow * DIM + k0 + 2 * half + 1];
    bb.x = WL[(k0 + 2 * half + 0) * DIM + tn * 16 + lrow];
    bb.y = WL[(k0 + 2 * half + 1) * DIM + tn * 16 + lrow];
    c = __builtin_amdgcn_wmma_f32_16x16x4_f32(false, a, false, bb,
                                              (short)0, c, false, false);
  }
  return c;
}

// Kernel 1: iteration 0, hop 1. One workgroup per drug b.
// Produces h1[b][m][d] = sigmoid((sv+agg) @ W + bias), m in [0,32).
__global__ __launch_bounds__(256) void kgcn_hop1(
    const int* __restrict__ drug, const int* __restrict__ adj_ent,
    const int* __restrict__ adj_rel, const float* __restrict__ ent_emb,
    const float* __restrict__ rel_emb, const float* __restrict__ W,
    const float* __restrict__ bias, float* __restrict__ h1) {
  __shared__ float relL[NREL * DIM];   // 8 KB: all relation embeddings
  __shared__ float WL[DIM * DIM];      // 4 KB
  __shared__ float bL[DIM];
  __shared__ float XL[32 * DIM];       // rows of (sv + agg) for the linear

  const int tid = threadIdx.x, wave = tid >> 5, lane = tid & 31;
  for (int i = tid; i < NREL * DIM; i += 256) relL[i] = rel_emb[i];
  for (int i = tid; i < DIM * DIM; i += 256) WL[i] = W[i];
  if (tid < DIM) bL[tid] = bias[tid];
  const int b = blockIdx.x;
  const int e0 = drug[b];
  __syncthreads();

  // 8 waves sweep the 32 rows; lane plays two roles: neighbor n, then dim d.
  for (int m = wave; m < 32; m += 8) {
    const int e1 = adj_ent[e0 * NN + m];
    const float sv = ent_emb[(long)e1 * DIM + lane];       // lane = d
    const int e2 = adj_ent[e1 * NN + lane];                // lane = n
    const int r  = adj_rel[e1 * NN + lane];
    __builtin_prefetch(&ent_emb[(long)e2 * DIM], 0, 1);    // global_prefetch_b8

    float score = 0.f;                                     // <sv, rel[r_n]>
#pragma unroll
    for (int d = 0; d < DIM; ++d)
      score = fmaf(__shfl(sv, d, 32), relL[r * DIM + d], score);

    const float mx = wave_max(score);                      // softmax over n
    float p = __expf(score - mx);
    p /= wave_sum(p);

    float acc = 0.f;                                       // lane = d again
#pragma unroll
    for (int n = 0; n < NN; ++n) {
      const float w = __shfl(p, n, 32);
      const int en = __shfl(e2, n, 32);
      acc = fmaf(w, ent_emb[(long)en * DIM + lane], acc);  // coalesced 128B row
    }
    XL[m * DIM + lane] = sv + acc;
  }
  __syncthreads();

  // 32x32 linear: 4 tiles of 16x16, one wave each (EXEC all ones per wave).
  if (wave < 4) {
    const int tm = wave >> 1, tn = wave & 1;
    const v8f c = wmma_x_w(&XL[tm * 16 * DIM], WL, tn, lane);
    const int lrow = lane & 15, half = lane >> 4;
    const int col = tn * 16 + lrow;
    const float bc = bL[col];
#pragma unroll
    for (int j = 0; j < 8; ++j) {
      const int row = tm * 16 + j + 8 * half;
      const float v = c[j] + bc;
      h1[((long)b * 32 + row) * DIM + col] = 1.f / (1.f + __expf(-v));
    }
  }
}

// Kernel 2: iter 0 hop 0 + iter 1 hop 0. 16 drugs per workgroup, one wave each.
__global__ __launch_bounds__(512) void kgcn_final(
    const int* __restrict__ drug, const int* __restrict__ adj_ent,
    const int* __restrict__ adj_rel, const float* __restrict__ ent_emb,
    const float* __restrict__ rel_emb, const float* __restrict__ W,
    const float* __restrict__ bias, const float* __restrict__ h1,
    float* __restrict__ out) {
  __shared__ float relL[NREL * DIM];
  __shared__ float WL[DIM * DIM];
  __shared__ float bL[DIM];
  __shared__ float XL[16 * DIM];
  __shared__ float H0[16 * DIM];

  const int tid = threadIdx.x, wave = tid >> 5, lane = tid & 31;
  for (int i = tid; i < NREL * DIM; i += 512) relL[i] = rel_emb[i];
  for (int i = tid; i < DIM * DIM; i += 512) WL[i] = W[i];
  if (tid < DIM) bL[tid] = bias[tid];
  const int b = blockIdx.x * 16 + wave;
  const int e0 = drug[b];
  const int en = adj_ent[e0 * NN + lane];   // lane = neighbor n
  const int rn = adj_rel[e0 * NN + lane];
  const float sv = ent_emb[(long)e0 * DIM + lane];  // lane = d
  __syncthreads();

  // --- iter 0, hop 0 ---
  float score = 0.f;
#pragma unroll
  for (int d = 0; d < DIM; ++d)
    score = fmaf(__shfl(sv, d, 32), relL[rn * DIM + d], score);
  float p = __expf(score - wave_max(score));
  p /= wave_sum(p);
  float acc = 0.f;
#pragma unroll
  for (int n = 0; n < NN; ++n) {
    const float w = __shfl(p, n, 32);
    const int e = __shfl(en, n, 32);
    acc = fmaf(w, ent_emb[(long)e * DIM + lane], acc);
  }
  XL[wave * DIM + lane] = sv + acc;
  __syncthreads();

  if (wave < 2) {  // 16x32 linear -> sigmoid -> H0
    const v8f c = wmma_x_w(XL, WL, wave, lane);
    const int lrow = lane & 15, half = lane >> 4;
    const int col = wave * 16 + lrow;
    const float bc = bL[col];
#pragma unroll
    for (int j = 0; j < 8; ++j) {
      const float v = c[j] + bc;
      H0[(j + 8 * half) * DIM + col] = 1.f / (1.f + __expf(-v));
    }
  }
  __syncthreads();

  // --- iter 1, hop 0: self = H0 row, neighbors = h1[b] rows, same relations ---
  float score2 = 0.f;
#pragma unroll
  for (int d = 0; d < DIM; ++d)
    score2 = fmaf(H0[wave * DIM + d], relL[rn * DIM + d], score2);
  float p2 = __expf(score2 - wave_max(score2));
  p2 /= wave_sum(p2);
  float acc2 = 0.f;
#pragma unroll
  for (int n = 0; n < NN; ++n) {
    const float w = __shfl(p2, n, 32);
    acc2 = fmaf(w, h1[((long)b * 32 + n) * DIM + lane], acc2);
  }
  __syncthreads();                       // all reads of XL/H0 phase done
  XL[wave * DIM + lane] = H0[wave * DIM + lane] + acc2;
  __syncthreads();

  if (wave < 2) {  // final 16x32 linear -> tanh -> out
    const v8f c = wmma_x_w(XL, WL, wave, lane);
    const int lrow = lane & 15, half = lane >> 4;
    const int col = wave * 16 + lrow;
    const float bc = bL[col];
#pragma unroll
    for (int j = 0; j < 8; ++j) {
      const int row = blockIdx.x * 16 + j + 8 * half;
      out[(long)row * DIM + col] = tanhf(c[j] + bc);
    }
  }
}

extern "C" void kernel_launch(void* const* d_in, const int* in_sizes, int n_in,
                              void* d_out, int out_size, void* d_ws, size_t ws_size,
                              hipStream_t stream) {
  const int*   drug    = (const int*)d_in[0];
  const int*   adj_ent = (const int*)d_in[1];
  const int*   adj_rel = (const int*)d_in[2];
  const float* ent_emb = (const float*)d_in[3];
  const float* rel_emb = (const float*)d_in[4];
  const float* W       = (const float*)d_in[5];
  const float* bias    = (const float*)d_in[6];
  float* out = (float*)d_out;
  float* h1  = (float*)d_ws;  // BATCH*32*32 floats = 4 MB scratch

  kgcn_hop1<<<BATCH, 256, 0, stream>>>(drug, adj_ent, adj_rel, ent_emb,
                                       rel_emb, W, bias, h1);
  kgcn_final<<<BATCH / 16, 512, 0, stream>>>(drug, adj_ent, adj_rel, ent_emb,
                                             rel_emb, W, bias, h1, out);
}